// GNN_27178553049852
// MI455X (gfx1250) — compile-verified
//
#include <hip/hip_runtime.h>

#define NN 50000
#define NE 1600000
#define HD 128
#define OUTD 16
#define NB 64

typedef __attribute__((ext_vector_type(2))) float v2f;
typedef __attribute__((ext_vector_type(8))) float v8f;

// ---------------------------------------------------------------------------
// Degree / normalization
// ---------------------------------------------------------------------------
__global__ void deg_init_kernel(float* deg) {
    int i = blockIdx.x * blockDim.x + threadIdx.x;
    if (i < NN) deg[i] = 1.0f;  // self-loop contribution
}

__global__ void deg_count_kernel(const int* __restrict__ dst, float* deg) {
    int e = blockIdx.x * blockDim.x + threadIdx.x;
    if (e < NE) atomicAdd(&deg[dst[e]], 1.0f);
}

__global__ void dinv_kernel(float* deg) {
    int i = blockIdx.x * blockDim.x + threadIdx.x;
    if (i < NN) {
        float d = deg[i];
        deg[i] = (d > 0.0f) ? rsqrtf(d) : 0.0f;  // in-place: deg -> dinv
    }
}

// ---------------------------------------------------------------------------
// GEMM: out[N x 128] = h[N x 128] @ W[128 x 128] using V_WMMA_F32_16X16X4_F32
// One block = 16 rows; 8 waves each own one 16-wide column tile.
// ---------------------------------------------------------------------------
__global__ void __launch_bounds__(256) gcn_gemm_wmma(const float* __restrict__ h,
                                                     const float* __restrict__ W,
                                                     float* __restrict__ out) {
    __shared__ float lw[HD * HD];  // 64 KB of the 320 KB WGP LDS pool
    {
        const float4* Wv = (const float4*)W;
        float4* lv = (float4*)lw;
        for (int i = threadIdx.x; i < HD * HD / 4; i += 256) lv[i] = Wv[i];
    }
    __syncthreads();

    const int wave  = threadIdx.x >> 5;
    const int lane  = threadIdx.x & 31;
    const int lhalf = lane >> 4;   // 0 -> lanes 0-15, 1 -> lanes 16-31
    const int lmod  = lane & 15;
    const int row0  = blockIdx.x * 16;
    const int col0  = wave * 16;

    v8f c = {};
    const float* arow = h + (size_t)(row0 + lmod) * HD + 2 * lhalf;

#pragma unroll
    for (int k0 = 0; k0 < HD; k0 += 4) {
        // A (16x4 f32): v0/v1 hold K = k0+2*lhalf, k0+2*lhalf+1 for row M=lmod
        v2f a = *(const v2f*)(arow + k0);
        // B (4x16 f32): mirrored striping, column N = col0+lmod
        v2f b;
        b.x = lw[(k0 + 2 * lhalf)     * HD + col0 + lmod];
        b.y = lw[(k0 + 2 * lhalf + 1) * HD + col0 + lmod];
        c = __builtin_amdgcn_wmma_f32_16x16x4_f32(false, a, false, b,
                                                  (short)0, c, false, false);
    }

#pragma unroll
    for (int j = 0; j < 8; ++j) {
        int r = row0 + j + 8 * lhalf;  // C/D layout: VGPR j -> M=j (lo) / j+8 (hi)
        out[(size_t)r * HD + col0 + lmod] = c[j];
    }
}

// ---------------------------------------------------------------------------
// agg[i] = hW[i] * dinv[i]^2   (self-loop message, also re-initializes agg)
// 32 lanes per node, float4 per lane.
// ---------------------------------------------------------------------------
__global__ void gcn_selfloop_init(const float* __restrict__ hw,
                                  const float* __restrict__ dinv,
                                  float* __restrict__ agg) {
    long long tid = (long long)blockIdx.x * blockDim.x + threadIdx.x;
    int i = (int)(tid >> 5);
    if (i >= NN) return;
    int f = ((int)tid & 31) * 4;
    float w = dinv[i];
    w = w * w;
    float4 v = *(const float4*)(hw + (size_t)i * HD + f);
    v.x *= w; v.y *= w; v.z *= w; v.w *= w;
    *(float4*)(agg + (size_t)i * HD + f) = v;
}

// ---------------------------------------------------------------------------
// Edge scatter: agg[dst] += hW[src] * dinv[src]*dinv[dst]
// One wave per edge, float4 per lane -> global_atomic_add_f32 x4.
// ---------------------------------------------------------------------------
__global__ void gcn_scatter(const float* __restrict__ hw,
                            const int* __restrict__ src,
                            const int* __restrict__ dst,
                            const float* __restrict__ dinv,
                            float* __restrict__ agg) {
    long long tid = (long long)blockIdx.x * blockDim.x + threadIdx.x;
    int e = (int)(tid >> 5);
    if (e >= NE) return;
    int f = ((int)tid & 31) * 4;
    int s = src[e], d = dst[e];
    float w = dinv[s] * dinv[d];
    float4 v = *(const float4*)(hw + (size_t)s * HD + f);
    float* p = agg + (size_t)d * HD + f;
    atomicAdd(p + 0, v.x * w);
    atomicAdd(p + 1, v.y * w);
    atomicAdd(p + 2, v.z * w);
    atomicAdd(p + 3, v.w * w);
}

// ---------------------------------------------------------------------------
// hout = relu(agg + b)
// ---------------------------------------------------------------------------
__global__ void gcn_relu_bias(const float* __restrict__ agg,
                              const float* __restrict__ b,
                              float* __restrict__ hout) {
    long long tid = (long long)blockIdx.x * blockDim.x + threadIdx.x;
    int i = (int)(tid >> 5);
    if (i >= NN) return;
    int f = ((int)tid & 31) * 4;
    float4 v = *(const float4*)(agg + (size_t)i * HD + f);
    v.x = fmaxf(v.x + b[f + 0], 0.0f);
    v.y = fmaxf(v.y + b[f + 1], 0.0f);
    v.z = fmaxf(v.z + b[f + 2], 0.0f);
    v.w = fmaxf(v.w + b[f + 3], 0.0f);
    *(float4*)(hout + (size_t)i * HD + f) = v;
}

// ---------------------------------------------------------------------------
// Pooling
// ---------------------------------------------------------------------------
__global__ void pool_zero_kernel(float* sums, float* cnts) {
    int i = blockIdx.x * blockDim.x + threadIdx.x;
    if (i < NB * HD) sums[i] = 0.0f;
    if (i < NB) cnts[i] = 0.0f;
}

__global__ void pool_accum_kernel(const float* __restrict__ h,
                                  const int* __restrict__ batch,
                                  float* __restrict__ sums) {
    long long tid = (long long)blockIdx.x * blockDim.x + threadIdx.x;
    int i = (int)(tid >> 5);
    if (i >= NN) return;
    int f = ((int)tid & 31) * 4;
    int g = batch[i];
    float4 v = *(const float4*)(h + (size_t)i * HD + f);
    float* p = sums + (size_t)g * HD + f;
    atomicAdd(p + 0, v.x);
    atomicAdd(p + 1, v.y);
    atomicAdd(p + 2, v.z);
    atomicAdd(p + 3, v.w);
}

__global__ void pool_count_kernel(const int* __restrict__ batch, float* cnts) {
    int i = blockIdx.x * blockDim.x + threadIdx.x;
    if (i < NN) atomicAdd(&cnts[batch[i]], 1.0f);
}

// out[g][o] = (sums[g]/max(cnt,1)) . lin_W[:,o] + lin_b[o]
__global__ void final_proj_kernel(const float* __restrict__ sums,
                                  const float* __restrict__ cnts,
                                  const float* __restrict__ linW,
                                  const float* __restrict__ linb,
                                  float* __restrict__ out) {
    int tid = blockIdx.x * blockDim.x + threadIdx.x;
    if (tid >= NB * OUTD) return;
    int g = tid / OUTD;
    int o = tid % OUTD;
    float inv = 1.0f / fmaxf(cnts[g], 1.0f);
    float acc = linb[o];
    for (int f = 0; f < HD; ++f)
        acc += sums[(size_t)g * HD + f] * inv * linW[(size_t)f * OUTD + o];
    out[(size_t)g * OUTD + o] = acc;
}

// ---------------------------------------------------------------------------
// Launch
// ---------------------------------------------------------------------------
extern "C" void kernel_launch(void* const* d_in, const int* in_sizes, int n_in,
                              void* d_out, int out_size, void* d_ws, size_t ws_size,
                              hipStream_t stream) {
    const float* x     = (const float*)d_in[0];
    const int*   eidx  = (const int*)d_in[1];
    const int*   batch = (const int*)d_in[2];
    const float* W0    = (const float*)d_in[4];
    const float* b0    = (const float*)d_in[5];
    const float* W1    = (const float*)d_in[6];
    const float* b1    = (const float*)d_in[7];
    const float* W2    = (const float*)d_in[8];
    const float* b2    = (const float*)d_in[9];
    const float* linW  = (const float*)d_in[10];
    const float* linb  = (const float*)d_in[11];
    float* out = (float*)d_out;

    const int* src = eidx;        // edge_index[0]
    const int* dst = eidx + NE;   // edge_index[1]

    float* ws   = (float*)d_ws;
    float* bufB = ws;                       // hW  : N*H
    float* bufC = bufB + (size_t)NN * HD;   // agg : N*H
    float* bufA = bufC + (size_t)NN * HD;   // h   : N*H
    float* dinv = bufA + (size_t)NN * HD;   // N
    float* sums = dinv + NN;                // NB*H
    float* cnts = sums + NB * HD;           // NB

    const int nodeBlk = (NN + 255) / 256;
    const int edgeBlk = (NE + 255) / 256;
    const int nodeF   = (NN * 32) / 256;    // 6250 (exact)
    const int edgeF   = (NE * 32) / 256;    // 200000 (exact)
    const int gemmBlk = NN / 16;            // 3125 (exact)

    // degrees -> dinv
    deg_init_kernel<<<nodeBlk, 256, 0, stream>>>(dinv);
    deg_count_kernel<<<edgeBlk, 256, 0, stream>>>(dst, dinv);
    dinv_kernel<<<nodeBlk, 256, 0, stream>>>(dinv);

    const float* hin = x;
    const float* Ws[3] = {W0, W1, W2};
    const float* bs[3] = {b0, b1, b2};
    for (int l = 0; l < 3; ++l) {
        gcn_gemm_wmma<<<gemmBlk, 256, 0, stream>>>(hin, Ws[l], bufB);
        gcn_selfloop_init<<<nodeF, 256, 0, stream>>>(bufB, dinv, bufC);
        gcn_scatter<<<edgeF, 256, 0, stream>>>(bufB, src, dst, dinv, bufC);
        gcn_relu_bias<<<nodeF, 256, 0, stream>>>(bufC, bs[l], bufA);
        hin = bufA;
    }

    // global mean pool + final linear
    pool_zero_kernel<<<(NB * HD + 255) / 256, 256, 0, stream>>>(sums, cnts);
    pool_accum_kernel<<<nodeF, 256, 0, stream>>>(bufA, batch, sums);
    pool_count_kernel<<<nodeBlk, 256, 0, stream>>>(batch, cnts);
    final_proj_kernel<<<(NB * OUTD + 255) / 256, 256, 0, stream>>>(sums, cnts, linW, linb, out);
}